// CausalSelectiveSelfAttentionForInference_60438779789955
// MI455X (gfx1250) — compile-verified
//
#include <hip/hip_runtime.h>
#include <hip/hip_bf16.h>

#define BB 2
#define TT 2048
#define CC 1024
#define NH 16
#define HD 64
#define C3 3072
#define BUDGET 408   /* memory_budget(409) - 1 */

typedef __attribute__((ext_vector_type(2))) float v2f;
typedef __attribute__((ext_vector_type(8))) float v8f;

__device__ __forceinline__ v8f wmma4(v2f a, v2f b, v8f c) {
  return __builtin_amdgcn_wmma_f32_16x16x4_f32(false, a, false, b, (short)0, c, false, false);
}

__device__ __forceinline__ float rowmax16(float x) {
  #pragma unroll
  for (int d = 1; d < 16; d <<= 1) x = fmaxf(x, __shfl_xor(x, d, 32));
  return x;
}
__device__ __forceinline__ float rowsum16(float x) {
  #pragma unroll
  for (int d = 1; d < 16; d <<= 1) x += __shfl_xor(x, d, 32);
  return x;
}

// ---------------------------------------------------------------------------
// Out[m][n] = sum_k A[m][k] * W[n][k] + bias[n]   (torch Linear)
// block = 128 threads (4 waves); wave computes 32 rows x 64 cols.
// Register double-buffered k-loop: next fragments in flight during WMMAs.
// ---------------------------------------------------------------------------
__global__ void gemm_wmma(const float* __restrict__ A, const float* __restrict__ W,
                          const float* __restrict__ bias, float* __restrict__ Out,
                          int M, int N, int K) {
  int wave = threadIdx.x >> 5;
  int lane = threadIdx.x & 31;
  int g  = lane >> 4;       // half-wave id
  int lm = lane & 15;
  int m0 = blockIdx.y * 32;
  int n0 = blockIdx.x * 256 + wave * 64;
  if (n0 >= N || m0 >= M) return;

  v8f acc[2][4];
  #pragma unroll
  for (int t = 0; t < 4; ++t) {
    float bv = bias[n0 + t * 16 + lm];
    #pragma unroll
    for (int mt = 0; mt < 2; ++mt)
      #pragma unroll
      for (int r = 0; r < 8; ++r) acc[mt][t][r] = bv;
  }
  const float* Ar[2];
  Ar[0] = A + (size_t)(m0 + lm) * K + 2 * g;        // A frag: M=lm, K=k0+2g+{0,1}
  Ar[1] = A + (size_t)(m0 + 16 + lm) * K + 2 * g;
  const float* Wp[4];
  #pragma unroll
  for (int t = 0; t < 4; ++t)
    Wp[t] = W + (size_t)(n0 + t * 16 + lm) * K + 2 * g;  // B[kk][n] = W[n][kk]

  v2f a[2], b[4], an[2], bn[4];
  #pragma unroll
  for (int mt = 0; mt < 2; ++mt) a[mt] = *(const v2f*)(Ar[mt]);
  #pragma unroll
  for (int t = 0; t < 4; ++t) b[t] = *(const v2f*)(Wp[t]);

  for (int k0 = 0; k0 < K; k0 += 4) {
    int k1 = (k0 + 4 < K) ? (k0 + 4) : k0;   // clamped prefetch (last iter reloads)
    an[0] = *(const v2f*)(Ar[0] + k1);
    an[1] = *(const v2f*)(Ar[1] + k1);
    #pragma unroll
    for (int t = 0; t < 4; ++t) bn[t] = *(const v2f*)(Wp[t] + k1);
    #pragma unroll
    for (int mt = 0; mt < 2; ++mt)
      #pragma unroll
      for (int t = 0; t < 4; ++t)
        acc[mt][t] = wmma4(a[mt], b[t], acc[mt][t]);
    a[0] = an[0]; a[1] = an[1];
    #pragma unroll
    for (int t = 0; t < 4; ++t) b[t] = bn[t];
  }
  #pragma unroll
  for (int mt = 0; mt < 2; ++mt)
    #pragma unroll
    for (int t = 0; t < 4; ++t)
      #pragma unroll
      for (int r = 0; r < 8; ++r)
        Out[(size_t)(m0 + mt * 16 + r + 8 * g) * N + n0 + t * 16 + lm] = acc[mt][t][r];
}

// ---------------------------------------------------------------------------
// Head-0 scores: S[b,i,j] = relu(q0_i . k0_j / 8)  with causal / col0 / diag zeroed.
// grid (T/16, T/16, B), block 32 (one wave per 16x16 tile).
// ---------------------------------------------------------------------------
__global__ void head0_scores(const float* __restrict__ qkv, float* __restrict__ S) {
  int jb = blockIdx.x, ib = blockIdx.y, b = blockIdx.z;
  int lane = threadIdx.x & 31, g = lane >> 4, lm = lane & 15;
  int i0 = ib * 16, j0 = jb * 16;
  if (jb > ib) {            // strictly above diagonal: all zeros
    #pragma unroll
    for (int r = 0; r < 8; ++r)
      S[((size_t)(b * TT + i0 + r + 8 * g)) * TT + j0 + lm] = 0.f;
    return;
  }
  const float* qrow = qkv + ((size_t)(b * TT) + i0 + lm) * C3;          // head0 q
  const float* krow = qkv + ((size_t)(b * TT) + j0 + lm) * C3 + CC;     // head0 k
  v2f aq[16], bk[16];
  #pragma unroll
  for (int ks = 0; ks < 16; ++ks) {
    aq[ks] = *(const v2f*)(qrow + ks * 4 + 2 * g);
    bk[ks] = *(const v2f*)(krow + ks * 4 + 2 * g);
  }
  v8f acc = {0.f, 0.f, 0.f, 0.f, 0.f, 0.f, 0.f, 0.f};
  #pragma unroll
  for (int ks = 0; ks < 16; ++ks) acc = wmma4(aq[ks], bk[ks], acc);
  #pragma unroll
  for (int r = 0; r < 8; ++r) {
    int i = i0 + r + 8 * g, j = j0 + lm;
    float s = acc[r] * 0.125f;                  // 1/sqrt(64)
    s = (j < i && j > 0) ? fmaxf(s, 0.f) : 0.f; // causal(j<=i), diag->0, col0->0
    S[((size_t)(b * TT + i)) * TT + j] = s;
  }
}

// ---------------------------------------------------------------------------
// In-place exclusive prefix over query rows: FF[i,j] = sum_{i'<i} S[i',j].
// One thread per (b, j); coalesced across j.
// ---------------------------------------------------------------------------
__global__ void ff_cumsum(float* __restrict__ S) {
  int j = blockIdx.x * blockDim.x + threadIdx.x;
  int b = blockIdx.y;
  float* col = S + (size_t)b * TT * TT + j;
  float carry = 0.f;
  #pragma unroll 8
  for (int i = 0; i < TT; ++i) {
    float v = col[(size_t)i * TT];
    col[(size_t)i * TT] = carry;
    carry += v;
  }
}

// ---------------------------------------------------------------------------
// Per row (b,i): radix-select 408th smallest FF among j<=i, emit keep bitmask
// (64 uint32 words per row). Stable index tie-break at threshold; diag forced.
// grid = B*T blocks, 256 threads.
// ---------------------------------------------------------------------------
__global__ void select_mask(const float* __restrict__ FF, unsigned* __restrict__ mask) {
  int row = blockIdx.x;
  int i = row & (TT - 1);
  const float* f = FF + (size_t)row * TT;
  unsigned* mrow = mask + (size_t)row * (TT / 32);
  int tid = threadIdx.x;

  if (i < BUDGET) {         // keep everything causal
    for (int w = tid; w < TT / 32; w += 256) {
      int jb = w * 32;
      unsigned m;
      if (jb + 31 <= i) m = 0xFFFFFFFFu;
      else if (jb > i)  m = 0u;
      else              m = (1u << (i - jb + 1)) - 1u;
      mrow[w] = m;
    }
    return;
  }

  __shared__ unsigned hist[256];
  __shared__ unsigned misc[4];      // [0]=prefix bits, [1]=k remaining, [2]=n_less
  __shared__ unsigned wave_tot[8];

  if (tid == 0) { misc[0] = 0u; misc[1] = BUDGET - 1; }
  unsigned prefix = 0, prefmask = 0;
  for (int pass = 0; pass < 4; ++pass) {
    int shift = 24 - 8 * pass;
    hist[tid] = 0u;
    __syncthreads();
    for (int j = tid; j <= i; j += 256) {
      unsigned u = __float_as_uint(f[j]);     // FF >= 0 -> bits order-preserving
      if ((u & prefmask) == prefix) atomicAdd(&hist[(u >> shift) & 255u], 1u);
    }
    __syncthreads();
    if (tid == 0) {
      unsigned kk = misc[1], cum = 0; int d = 0;
      for (; d < 256; ++d) { unsigned h = hist[d]; if (cum + h > kk) break; cum += h; }
      misc[1] = kk - cum;
      misc[0] |= ((unsigned)d << shift);
    }
    __syncthreads();
    prefix = misc[0];
    prefmask |= (255u << shift);
  }
  unsigned theta = prefix;

  if (tid == 0) misc[2] = 0u;
  __syncthreads();
  unsigned cl = 0;
  for (int j = tid; j <= i; j += 256)
    if (__float_as_uint(f[j]) < theta) cl++;
  atomicAdd(&misc[2], cl);
  __syncthreads();
  int ntake = BUDGET - (int)misc[2];   // #equal-to-theta keys to keep (>=1)

  int lane = tid & 31, wv = tid >> 5;
  unsigned below = (1u << lane) - 1u;
  unsigned carry = 0;
  for (int base = 0; base < TT; base += 256) {
    int j = base + tid;
    unsigned u = (j <= i) ? __float_as_uint(f[j]) : 0xFFFFFFFFu;
    bool eq = (j <= i) && (u == theta);
    unsigned bal = (unsigned)__ballot(eq);
    if (lane == 0) wave_tot[wv] = __popc(bal);
    __syncthreads();
    unsigned pre = carry + __popc(bal & below);
    for (int w2 = 0; w2 < wv; ++w2) pre += wave_tot[w2];
    bool keep = (j <= i) && ((u < theta) || (eq && (int)pre < ntake));
    keep = keep || (j == i);
    unsigned kb = (unsigned)__ballot(keep);
    if (lane == 0) mrow[(base + wv * 32) >> 5] = kb;
    unsigned tot = 0;
    #pragma unroll
    for (int w2 = 0; w2 < 8; ++w2) tot += wave_tot[w2];
    carry += tot;
    __syncthreads();
  }
}

// ---------------------------------------------------------------------------
// Flash-style masked attention: one wave per (b, h, 16-row query tile).
// s = qk/8 - FF (kept) else -1e30; online softmax; P.V via WMMA.
// All fragment loads batched ahead of the WMMA chains.
// ---------------------------------------------------------------------------
__global__ void attn(const float* __restrict__ qkv, const float* __restrict__ FF,
                     const unsigned* __restrict__ mask, float* __restrict__ Y) {
  int ib = blockIdx.x, h = blockIdx.y, b = blockIdx.z;
  int lane = threadIdx.x & 31, g = lane >> 4, lm = lane & 15;
  int i0 = ib * 16;
  __shared__ float pt[16 * 17];

  const float* qb = qkv + ((size_t)(b * TT)) * C3 + h * HD;
  const float* kb = qb + CC;
  const float* vb = qb + 2 * CC;

  v2f aq[16];
  #pragma unroll
  for (int ks = 0; ks < 16; ++ks)
    aq[ks] = *(const v2f*)(qb + (size_t)(i0 + lm) * C3 + ks * 4 + 2 * g);

  float mrow[8], lrow[8];
  v8f o[4];
  #pragma unroll
  for (int t = 0; t < 4; ++t)
    #pragma unroll
    for (int r = 0; r < 8; ++r) o[t][r] = 0.f;
  #pragma unroll
  for (int r = 0; r < 8; ++r) { mrow[r] = -1e30f; lrow[r] = 0.f; }

  for (int jb = 0; jb <= ib; ++jb) {
    int j0 = jb * 16;
    // ---- batch K fragments, then QK^T WMMA chain
    v2f bk[16];
    #pragma unroll
    for (int ks = 0; ks < 16; ++ks)
      bk[ks] = *(const v2f*)(kb + (size_t)(j0 + lm) * C3 + ks * 4 + 2 * g);
    v8f acc = {0.f, 0.f, 0.f, 0.f, 0.f, 0.f, 0.f, 0.f};
    #pragma unroll
    for (int ks = 0; ks < 16; ++ks) acc = wmma4(aq[ks], bk[ks], acc);

    // ---- batch FF + mask loads
    float ffv[8]; unsigned mwv[8];
    #pragma unroll
    for (int r = 0; r < 8; ++r) {
      int i = i0 + r + 8 * g;
      ffv[r] = FF[((size_t)(b * TT + i)) * TT + j0 + lm];
      mwv[r] = mask[((size_t)(b * TT + i)) * (TT / 32) + (j0 >> 5)];
    }
    // ---- online softmax update (per C-layout row slot)
    float p[8];
    #pragma unroll
    for (int r = 0; r < 8; ++r) {
      int i = i0 + r + 8 * g, j = j0 + lm;
      bool keep = ((mwv[r] >> (j & 31)) & 1u) && (j <= i);
      float s = keep ? (acc[r] * 0.125f - ffv[r]) : -1e30f;
      float bm = rowmax16(s);
      float mnew = fmaxf(mrow[r], bm);
      float scale = __expf(mrow[r] - mnew);
      float pr = __expf(s - mnew);
      lrow[r] = lrow[r] * scale + rowsum16(pr);
      mrow[r] = mnew;
      p[r] = pr;
      #pragma unroll
      for (int t = 0; t < 4; ++t) o[t][r] *= scale;
    }
    // ---- stage P through LDS to re-layout C->A fragments
    __syncthreads();
    #pragma unroll
    for (int r = 0; r < 8; ++r) pt[(r + 8 * g) * 17 + lm] = p[r];
    __syncthreads();
    v2f ap[4];
    #pragma unroll
    for (int kt = 0; kt < 4; ++kt) {
      ap[kt].x = pt[lm * 17 + kt * 4 + 2 * g];
      ap[kt].y = pt[lm * 17 + kt * 4 + 2 * g + 1];
    }
    // ---- batch V fragments, then P.V WMMA chain
    v2f bv[4][4];
    #pragma unroll
    for (int kt = 0; kt < 4; ++kt)
      #pragma unroll
      for (int t = 0; t < 4; ++t) {
        bv[kt][t].x = vb[(size_t)(j0 + kt * 4 + 2 * g) * C3 + t * 16 + lm];
        bv[kt][t].y = vb[(size_t)(j0 + kt * 4 + 2 * g + 1) * C3 + t * 16 + lm];
      }
    #pragma unroll
    for (int kt = 0; kt < 4; ++kt)
      #pragma unroll
      for (int t = 0; t < 4; ++t)
        o[t] = wmma4(ap[kt], bv[kt][t], o[t]);
  }
  #pragma unroll
  for (int r = 0; r < 8; ++r) {
    float inv = 1.f / lrow[r];
    int i = i0 + r + 8 * g;
    #pragma unroll
    for (int t = 0; t < 4; ++t)
      Y[((size_t)(b * TT + i)) * CC + h * HD + t * 16 + lm] = o[t][r] * inv;
  }
}

// ---------------------------------------------------------------------------
extern "C" void kernel_launch(void* const* d_in, const int* in_sizes, int n_in,
                              void* d_out, int out_size, void* d_ws, size_t ws_size,
                              hipStream_t stream) {
  const float* x      = (const float*)d_in[0];
  const float* w_attn = (const float*)d_in[1];
  const float* b_attn = (const float*)d_in[2];
  const float* w_proj = (const float*)d_in[3];
  const float* b_proj = (const float*)d_in[4];
  float* out = (float*)d_out;

  char* ws = (char*)d_ws;
  const size_t QKV_BYTES = (size_t)BB * TT * C3 * 4;        // 50.3 MB
  const size_t FF_BYTES  = (size_t)BB * TT * TT * 4;        // 33.5 MB
  const size_t MSK_BYTES = (size_t)BB * TT * (TT / 32) * 4; //  1.0 MB
  float*    qkv = (float*)ws;
  float*    FF  = (float*)(ws + QKV_BYTES);
  unsigned* msk = (unsigned*)(ws + QKV_BYTES + FF_BYTES);
  float*    Y   = (float*)(ws + QKV_BYTES + FF_BYTES + MSK_BYTES);

  // 1. qkv = x @ w_attn.T + b_attn   [4096 x 3072]
  gemm_wmma<<<dim3(C3 / 256, (BB * TT) / 32), 128, 0, stream>>>(
      x, w_attn, b_attn, qkv, BB * TT, C3, CC);
  // 2. head-0 masked relu scores -> S (in FF buffer)
  head0_scores<<<dim3(TT / 16, TT / 16, BB), 32, 0, stream>>>(qkv, FF);
  // 3. exclusive cumsum over query rows -> FF (in place)
  ff_cumsum<<<dim3(TT / 256, BB), 256, 0, stream>>>(FF);
  // 4. per-row bottom-408 FF selection -> keep bitmask
  select_mask<<<BB * TT, 256, 0, stream>>>(FF, msk);
  // 5. masked flash attention -> Y [4096 x 1024]
  attn<<<dim3(TT / 16, NH, BB), 32, 0, stream>>>(qkv, FF, msk, Y);
  // 6. out = Y @ w_proj.T + b_proj
  gemm_wmma<<<dim3(CC / 256, (BB * TT) / 32), 128, 0, stream>>>(
      Y, w_proj, b_proj, out, BB * TT, CC, CC);
}